// GNN_64020782514491
// MI455X (gfx1250) — compile-verified
//
#include <hip/hip_runtime.h>
#include <hip/hip_bf16.h>

// GCN 3-layer forward for MI455X (gfx1250), wave32.
//   h   = x @ W                      (fp32 WMMA 16x16x4)
//   agg = segment_sum(coef * h[src]) (edge scatter, global_atomic_add_f32)
//   out = relu(agg + dinv^2*h + b)   (self-loop folded into finalize)

#define NODES 50000
#define HDIM  128

typedef __attribute__((ext_vector_type(2))) float v2f;
typedef __attribute__((ext_vector_type(8))) float v8f;

// ---------------------------------------------------------------------------
// GEMM: Hout[N x 128] = X[N x 128] @ W[128 x 128], fp32 WMMA.
// One block = 256 threads = 8 waves. Block handles 16 rows x 128 cols.
// Wave w -> 16x16 output tile at colBase = 16*w.  K looped 0..128 step 4.
// ---------------------------------------------------------------------------
__global__ __launch_bounds__(256) void gemm128_wmma(const float* __restrict__ X,
                                                    const float* __restrict__ W,
                                                    float* __restrict__ Hout) {
    // W staged whole (64 KB), X tile padded to 130 floats/row to avoid the
    // stride-128 (bank-aligned) LDS conflict on A-fragment reads.
    __shared__ float Wl[128 * 128];
    __shared__ float Xl[16 * 130];

    const int tid = threadIdx.x;
    const int rowBase = blockIdx.x * 16;

    // cooperative load of W via b128
    {
        const float4* W4 = (const float4*)W;
        float4* Wl4 = (float4*)Wl;
        for (int i = tid; i < 128 * 128 / 4; i += 256) Wl4[i] = W4[i];
    }
    // cooperative load of the 16x128 X tile into padded LDS rows
    {
        const float* Xt = X + (size_t)rowBase * HDIM;
        for (int i = tid; i < 16 * 128; i += 256) {
            int r = i >> 7, c = i & 127;
            Xl[r * 130 + c] = Xt[i];
        }
    }
    __syncthreads();

    const int lane = tid & 31;
    const int wave = tid >> 5;
    const int r16  = lane & 15;   // lane-in-halfwave: A row M / B-D col N
    const int hi   = lane >> 4;   // half-wave select (K offset +2 / M offset +8)
    const int colBase = wave * 16;

    v8f acc = {};
    #pragma unroll 4
    for (int k = 0; k < 128; k += 4) {
        const int kk = k + hi * 2;
        // A 16x4 layout: lanes 0-15 -> (M=lane, K=k..k+1), lanes 16-31 -> K=k+2..k+3
        v2f a;
        a[0] = Xl[r16 * 130 + kk];
        a[1] = Xl[r16 * 130 + kk + 1];
        // B 4x16 layout (mirror of A, transposed): lane holds N=r16,
        // VGPR j -> K = kk + j
        v2f b;
        b[0] = Wl[(kk + 0) * 128 + colBase + r16];
        b[1] = Wl[(kk + 1) * 128 + colBase + r16];
        acc = __builtin_amdgcn_wmma_f32_16x16x4_f32(
            /*neg_a=*/false, a, /*neg_b=*/false, b,
            /*c_mod=*/(short)0, acc, /*reuse_a=*/false, /*reuse_b=*/false);
    }

    // D layout: VGPR r, lanes 0-15 -> (M=r, N=lane); lanes 16-31 -> (M=r+8)
    float* out = Hout + (size_t)rowBase * HDIM + colBase + r16;
    #pragma unroll
    for (int r = 0; r < 8; ++r) {
        out[(size_t)(r + hi * 8) * HDIM] = acc[r];
    }
}

// ---------------------------------------------------------------------------
// Degree / normalization
// ---------------------------------------------------------------------------
__global__ void init_deg(float* deg, int n) {
    int i = blockIdx.x * blockDim.x + threadIdx.x;
    if (i < n) deg[i] = 1.0f;  // self-loop contributes 1
}

__global__ void accum_deg(const int* __restrict__ dst, float* deg, int e) {
    int i = blockIdx.x * blockDim.x + threadIdx.x;
    if (i < e) atomicAdd(&deg[dst[i]], 1.0f);
}

__global__ void rsqrt_deg(float* deg, int n) {
    int i = blockIdx.x * blockDim.x + threadIdx.x;
    if (i < n) deg[i] = __frsqrt_rn(deg[i]);  // deg >= 1 always
}

// ---------------------------------------------------------------------------
// Zero the accumulator
// ---------------------------------------------------------------------------
__global__ void zero_f4(float4* p, int n4) {
    int i = blockIdx.x * blockDim.x + threadIdx.x;
    if (i < n4) p[i] = make_float4(0.f, 0.f, 0.f, 0.f);
}

// ---------------------------------------------------------------------------
// Edge aggregation: one wave (32 lanes) per edge, float4 per lane (128 feats).
// agg[dst] += dinv[src]*dinv[dst] * h[src]
// ---------------------------------------------------------------------------
__global__ __launch_bounds__(256) void aggregate(const float* __restrict__ h,
                                                 const int* __restrict__ src,
                                                 const int* __restrict__ dst,
                                                 const float* __restrict__ dinv,
                                                 float* __restrict__ agg, int e) {
    int g = blockIdx.x * blockDim.x + threadIdx.x;
    int edge = g >> 5;
    int lane = g & 31;
    if (edge >= e) return;
    int s = src[edge];
    int d = dst[edge];
    float c = dinv[s] * dinv[d];
    const float4 v = ((const float4*)(h + (size_t)s * HDIM))[lane];
    float* a = agg + (size_t)d * HDIM + lane * 4;
    atomicAdd(a + 0, c * v.x);
    atomicAdd(a + 1, c * v.y);
    atomicAdd(a + 2, c * v.z);
    atomicAdd(a + 3, c * v.w);
}

// ---------------------------------------------------------------------------
// Finalize: out = (agg + dinv[i]^2 * h + b), optional ReLU. float4 per thread.
// ---------------------------------------------------------------------------
__global__ __launch_bounds__(256) void finalize(const float* __restrict__ agg,
                                                const float* __restrict__ h,
                                                const float* __restrict__ dinv,
                                                const float* __restrict__ bias,
                                                float* __restrict__ out,
                                                int n, int relu) {
    int i = blockIdx.x * blockDim.x + threadIdx.x;  // over N*H/4
    if (i >= n * (HDIM / 4)) return;
    int node = i / (HDIM / 4);
    int c4   = i % (HDIM / 4);
    float sl = dinv[node] * dinv[node];
    float4 a = ((const float4*)agg)[i];
    float4 v = ((const float4*)h)[i];
    const float4 b = ((const float4*)bias)[c4];
    a.x += sl * v.x + b.x;
    a.y += sl * v.y + b.y;
    a.z += sl * v.z + b.z;
    a.w += sl * v.w + b.w;
    if (relu) {
        a.x = fmaxf(a.x, 0.f);
        a.y = fmaxf(a.y, 0.f);
        a.z = fmaxf(a.z, 0.f);
        a.w = fmaxf(a.w, 0.f);
    }
    ((float4*)out)[i] = a;
}

// ---------------------------------------------------------------------------
extern "C" void kernel_launch(void* const* d_in, const int* in_sizes, int n_in,
                              void* d_out, int out_size, void* d_ws, size_t ws_size,
                              hipStream_t stream) {
    (void)n_in; (void)out_size; (void)ws_size;

    const float* x  = (const float*)d_in[0];
    const int*   ei = (const int*)d_in[1];
    const float* W1 = (const float*)d_in[2];
    const float* b1 = (const float*)d_in[3];
    const float* W2 = (const float*)d_in[4];
    const float* b2 = (const float*)d_in[5];
    const float* W3 = (const float*)d_in[6];
    const float* b3 = (const float*)d_in[7];

    const int N = in_sizes[0] / HDIM;   // 50000
    const int E = in_sizes[1] / 2;      // 800000
    const int* src = ei;
    const int* dst = ei + E;

    float* ws   = (float*)d_ws;
    float* h    = ws;
    float* agg  = ws + (size_t)N * HDIM;
    float* xbuf = ws + 2 * (size_t)N * HDIM;
    float* dinv = ws + 3 * (size_t)N * HDIM;
    float* out  = (float*)d_out;

    const int nh4 = N * (HDIM / 4);
    dim3 blk(256);
    dim3 gN((N + 255) / 256);
    dim3 gE((E + 255) / 256);
    dim3 gEdge((E + 7) / 8);          // 8 edges (waves) per block
    dim3 gNH4((nh4 + 255) / 256);
    dim3 gGemm(N / 16);               // 50000 = 3125 * 16 exactly

    // --- normalization coefficients ---
    init_deg<<<gN, blk, 0, stream>>>(dinv, N);
    accum_deg<<<gE, blk, 0, stream>>>(dst, dinv, E);
    rsqrt_deg<<<gN, blk, 0, stream>>>(dinv, N);

    // --- layer 1 ---
    gemm128_wmma<<<gGemm, blk, 0, stream>>>(x, W1, h);
    zero_f4<<<gNH4, blk, 0, stream>>>((float4*)agg, nh4);
    aggregate<<<gEdge, blk, 0, stream>>>(h, src, dst, dinv, agg, E);
    finalize<<<gNH4, blk, 0, stream>>>(agg, h, dinv, b1, xbuf, N, 1);

    // --- layer 2 (output overwrites xbuf: finalize never reads layer input) ---
    gemm128_wmma<<<gGemm, blk, 0, stream>>>(xbuf, W2, h);
    zero_f4<<<gNH4, blk, 0, stream>>>((float4*)agg, nh4);
    aggregate<<<gEdge, blk, 0, stream>>>(h, src, dst, dinv, agg, E);
    finalize<<<gNH4, blk, 0, stream>>>(agg, h, dinv, b2, xbuf, N, 1);

    // --- layer 3 (no ReLU, write d_out) ---
    gemm128_wmma<<<gGemm, blk, 0, stream>>>(xbuf, W3, h);
    zero_f4<<<gNH4, blk, 0, stream>>>((float4*)agg, nh4);
    aggregate<<<gEdge, blk, 0, stream>>>(h, src, dst, dinv, agg, E);
    finalize<<<gNH4, blk, 0, stream>>>(agg, h, dinv, b3, out, N, 0);
}